// Encoder_21964462752332
// MI455X (gfx1250) — compile-verified
//
#include <hip/hip_runtime.h>
#include <math.h>

typedef __attribute__((ext_vector_type(2))) float v2f;
typedef __attribute__((ext_vector_type(8))) float v8f;

#define IMG_H 64
#define IMG_W 64
#define OH 31
#define OW 31
#define PPI (OH * OW)           /* 961 patches per image */
#define TILES_PER_BLOCK 8       /* 8 waves per 256-thread block, 1 tile per wave */

// ---------------------------------------------------------------------------
// Kernel 1: build B (16x16) where B[k][n] = U[n][k] for n<4 else 0.
// U = product of the MPS gates (2 layers x 3 blocks of RY,RY,CNOT) acting on
// the 4 data wires; wire b corresponds to bit mask (8 >> b) of the 16-index.
// 16 threads, thread c owns column c of U in registers.
// ---------------------------------------------------------------------------
__global__ void build_proj(const float* __restrict__ w, float* __restrict__ Bmat) {
    const int c = threadIdx.x;
    if (c >= 16) return;
    float u[16];
#pragma unroll
    for (int r = 0; r < 16; ++r) u[r] = (r == c) ? 1.0f : 0.0f;

#pragma unroll
    for (int l = 0; l < 2; ++l) {
#pragma unroll
        for (int b = 0; b < 3; ++b) {
            const int   m0  = 8 >> b;       // wire b
            const int   m1  = 4 >> b;       // wire b+1
            const float th0 = w[(l * 3 + b) * 2 + 0];
            const float th1 = w[(l * 3 + b) * 2 + 1];
            {   // RY(th0) on wire b  (left-multiply onto U)
                const float ct = cosf(0.5f * th0), st = sinf(0.5f * th0);
#pragma unroll
                for (int r = 0; r < 16; ++r) {
                    if (r & m0) continue;
                    const float a0 = u[r], a1 = u[r | m0];
                    u[r]      = ct * a0 - st * a1;
                    u[r | m0] = st * a0 + ct * a1;
                }
            }
            {   // RY(th1) on wire b+1
                const float ct = cosf(0.5f * th1), st = sinf(0.5f * th1);
#pragma unroll
                for (int r = 0; r < 16; ++r) {
                    if (r & m1) continue;
                    const float a0 = u[r], a1 = u[r | m1];
                    u[r]      = ct * a0 - st * a1;
                    u[r | m1] = st * a0 + ct * a1;
                }
            }
            // CNOT: control wire b, target wire b+1 -> swap rows (r, r|m1) for control=1
#pragma unroll
            for (int r = 0; r < 16; ++r) {
                if ((r & m0) && !(r & m1)) {
                    const float t = u[r];
                    u[r]      = u[r | m1];
                    u[r | m1] = t;
                }
            }
        }
    }
    // B[k][n] = U[n][k] for n<4 (swap-test needs only ψ_0..ψ_3), else 0.
#pragma unroll
    for (int n = 0; n < 16; ++n)
        Bmat[c * 16 + n] = (n < 4) ? u[n] : 0.0f;
}

// ---------------------------------------------------------------------------
// Kernel 2: one wave32 per tile of 16 patches.
// out[p] = (1 + sum_{n<4} (U amp_p)_n^2) / 2 via fp32 WMMA (4 x 16x16x4 chain)
// ---------------------------------------------------------------------------
__global__ __launch_bounds__(256) void swap_test_kernel(
    const float* __restrict__ img, const float* __restrict__ Bmat,
    float* __restrict__ out, int n_out) {

    const int lane = threadIdx.x & 31;
    const int wave = threadIdx.x >> 5;
    const int tile = blockIdx.x * TILES_PER_BLOCK + wave;
    if (tile * 16 >= n_out) return;            // wave-uniform: EXEC stays all-ones

    const int m = lane & 15;                   // A-matrix row (patch in tile)
    const int h = lane >> 4;                   // lane half -> which K pair

    // patch index -> (image, oy, ox)
    const int p   = tile * 16 + m;
    const int b   = p / PPI;
    const int rem = p - b * PPI;
    const int oy  = rem / OW;
    const int ox  = rem - oy * OW;

    // A layout (32-bit 16x4, chained over 4 K-chunks): lane holds
    // amp[m][4*kk + 2*h + v], v=0,1  -> two adjacent pixels of image row kk.
    const float* src = img + ((b * IMG_H + oy * 2) * IMG_W + ox * 2 + 2 * h);
    float e[8];
#pragma unroll
    for (int kk = 0; kk < 4; ++kk) {
        const float2 t = *reinterpret_cast<const float2*>(src + kk * IMG_W);
        e[2 * kk + 0] = t.x;
        e[2 * kk + 1] = t.y;
    }

    // normalization: this lane has 8 of 16 elements; partner lane (L^16) has the rest
    float n2 = 0.0f;
#pragma unroll
    for (int i = 0; i < 8; ++i) n2 += e[i] * e[i];
    n2 += __shfl_xor(n2, 16, 32);
    const float nrm = sqrtf(n2);
    if (nrm > 0.0f) {
        const float s = 1.0f / fmaxf(nrm, 1e-12f);
#pragma unroll
        for (int i = 0; i < 8; ++i) e[i] *= s;
    } else {
        // amp = e0 (element 0 lives at kk=0,h=0,v=0)
#pragma unroll
        for (int i = 0; i < 8; ++i) e[i] = 0.0f;
        if (h == 0) e[0] = 1.0f;
    }

    // B fragments: mirrored layout, lane holds B[4*kk + 2*h + v][n], n = lane&15
    const int n = m;
    float bf[8];
#pragma unroll
    for (int kk = 0; kk < 4; ++kk) {
        bf[2 * kk + 0] = Bmat[(4 * kk + 2 * h + 0) * 16 + n];
        bf[2 * kk + 1] = Bmat[(4 * kk + 2 * h + 1) * 16 + n];
    }

    // D[m][n] = sum_k amp[m][k] * B[k][n] = psi_m[n]  (cols n>=4 are zero)
    v8f acc = {};
#pragma unroll
    for (int kk = 0; kk < 4; ++kk) {
        v2f av; av[0] = e[2 * kk]; av[1] = e[2 * kk + 1];
        v2f bv; bv[0] = bf[2 * kk]; bv[1] = bf[2 * kk + 1];
        acc = __builtin_amdgcn_wmma_f32_16x16x4_f32(
            /*neg_a=*/false, av, /*neg_b=*/false, bv,
            /*c_mod=*/(short)0, acc, /*reuse_a=*/false, /*reuse_b=*/false);
    }

    // sum of squares over N (all 16 lanes of this half); rows m = r + 8*h
    float s[8];
#pragma unroll
    for (int r = 0; r < 8; ++r) s[r] = acc[r] * acc[r];
#pragma unroll
    for (int mask = 1; mask < 16; mask <<= 1) {
#pragma unroll
        for (int r = 0; r < 8; ++r) s[r] += __shfl_xor(s[r], mask, 32);
    }

    if (m == 0) {
        float* dst = out + tile * 16 + 8 * h;   // 32B-aligned
        const float4 o0 = make_float4((1.0f + s[0]) * 0.5f, (1.0f + s[1]) * 0.5f,
                                      (1.0f + s[2]) * 0.5f, (1.0f + s[3]) * 0.5f);
        const float4 o1 = make_float4((1.0f + s[4]) * 0.5f, (1.0f + s[5]) * 0.5f,
                                      (1.0f + s[6]) * 0.5f, (1.0f + s[7]) * 0.5f);
        *reinterpret_cast<float4*>(dst)     = o0;
        *reinterpret_cast<float4*>(dst + 4) = o1;
    }
}

extern "C" void kernel_launch(void* const* d_in, const int* in_sizes, int n_in,
                              void* d_out, int out_size, void* d_ws, size_t ws_size,
                              hipStream_t stream) {
    const float* img  = (const float*)d_in[0];   // (256,1,64,64) f32
    const float* wmps = (const float*)d_in[1];   // (2,3,2) f32
    float* out  = (float*)d_out;                 // (256,1,961) f32
    float* Bmat = (float*)d_ws;                  // 256 floats scratch

    build_proj<<<1, 16, 0, stream>>>(wmps, Bmat);

    const int tiles  = out_size / 16;                              // 15376
    const int blocks = (tiles + TILES_PER_BLOCK - 1) / TILES_PER_BLOCK; // 1922
    swap_test_kernel<<<blocks, 256, 0, stream>>>(img, Bmat, out, out_size);
}